// GraphNetwork_31988916420711
// MI455X (gfx1250) — compile-verified
//
#include <hip/hip_runtime.h>
#include <hip/hip_bf16.h>

// ---------------------------------------------------------------------------
// HypergraphConv (attention) for MI455X / gfx1250, wave32.
//   he       = hyperedge_attr @ weight              (WMMA f32 16x16x4 GEMM)
//   alpha    = leaky(x.att0[gather] + he.att1[gather], 0.2)
//   softmax over node segments, degree counts
//   edge_out = segsum_edge( Binv*alpha * x[node] )
//   node_out = segsum_node( Dinv*alpha * edge_out[edge] )
//   out      = leaky(node_out + bias, 0.01)
// ---------------------------------------------------------------------------

typedef __attribute__((ext_vector_type(2))) float v2f;
typedef __attribute__((ext_vector_type(8))) float v8f;

#define FDIM 256

__device__ __forceinline__ void atomicMaxF32(float* addr, float val) {
    // classic bit-trick: works with -inf init, mixed signs
    if (val >= 0.0f)
        atomicMax((int*)addr, __float_as_int(val));
    else
        atomicMin((unsigned int*)addr, __float_as_uint(val));
}

// ---------------------------------------------------------------------------
// 0) init: zero d_out accumulator, edge_out, denom, D, B; amax = -inf
// ---------------------------------------------------------------------------
__global__ __launch_bounds__(256)
void init_kernel(float* __restrict__ out_acc, float* __restrict__ edge_out,
                 float* __restrict__ amax, float* __restrict__ denom,
                 float* __restrict__ Dcnt, float* __restrict__ Bcnt,
                 long long nf, long long mf, int n_nodes, int n_edges) {
    long long stride = (long long)gridDim.x * blockDim.x;
    for (long long i = (long long)blockIdx.x * blockDim.x + threadIdx.x;
         i < nf; i += stride) {
        out_acc[i] = 0.0f;
        if (i < mf) edge_out[i] = 0.0f;
        if (i < n_nodes) { amax[i] = -INFINITY; denom[i] = 0.0f; Dcnt[i] = 0.0f; }
        if (i < n_edges) { Bcnt[i] = 0.0f; }
    }
}

// ---------------------------------------------------------------------------
// 1) he = A(MxF) @ W(FxF) with V_WMMA_F32_16X16X4_F32, one wave per 16x16 tile
//    A layout (16x4 f32): lanes 0-15 -> M=lane, v0=K0 v1=K1; lanes 16-31 K2,K3
//    B layout (4x16 f32): lanes give N; lanes 0-15 v0=K0 v1=K1; 16-31 K2,K3
//    C/D layout (16x16 f32): vgpr r -> M = r + (lane<16 ? 0 : 8), N = lane&15
// ---------------------------------------------------------------------------
__global__ __launch_bounds__(256)
void he_gemm_kernel(const float* __restrict__ A, const float* __restrict__ W,
                    float* __restrict__ he, int mrows) {
    const int lane = threadIdx.x & 31;
    const int wave = threadIdx.x >> 5;
    const int tilesN = FDIM / 16;                 // 16
    const int tile = blockIdx.x * 8 + wave;
    const int tM = tile / tilesN;
    const int tN = tile % tilesN;

    const int row   = tM * 16 + (lane & 15);
    const int col   = tN * 16 + (lane & 15);
    const int khalf = (lane >> 4) * 2;            // 0 or 2
    const bool fullTile = (tM * 16 + 16 <= mrows); // wave-uniform
    const bool rowOk = (row < mrows);
    const float* arow = A + (size_t)(rowOk ? row : 0) * FDIM;

    v8f c = {};
    if (fullTile) {
        // hot path: no per-lane guards at all
        #pragma unroll 4
        for (int kk = 0; kk < FDIM; kk += 4) {
            const int k = kk + khalf;
            v2f a, b;
            a.x = arow[k];
            a.y = arow[k + 1];
            b.x = W[(size_t)k * FDIM + col];
            b.y = W[(size_t)(k + 1) * FDIM + col];
            c = __builtin_amdgcn_wmma_f32_16x16x4_f32(
                    false, a, false, b, (short)0, c, false, false);
        }
        const int crowbase = tM * 16 + (lane >> 4) * 8;
        #pragma unroll
        for (int r = 0; r < 8; ++r)
            he[(size_t)(crowbase + r) * FDIM + col] = c[r];
    } else {
        // boundary tiles (only the last tile row)
        #pragma unroll 4
        for (int kk = 0; kk < FDIM; kk += 4) {
            const int k = kk + khalf;
            float ax = arow[k];
            float ay = arow[k + 1];
            v2f a, b;
            a.x = rowOk ? ax : 0.0f;
            a.y = rowOk ? ay : 0.0f;
            b.x = W[(size_t)k * FDIM + col];
            b.y = W[(size_t)(k + 1) * FDIM + col];
            c = __builtin_amdgcn_wmma_f32_16x16x4_f32(
                    false, a, false, b, (short)0, c, false, false);
        }
        const int crowbase = tM * 16 + (lane >> 4) * 8;
        #pragma unroll
        for (int r = 0; r < 8; ++r) {
            const int crow = crowbase + r;
            if (crow < mrows) he[(size_t)crow * FDIM + col] = c[r];
        }
    }
}

// ---------------------------------------------------------------------------
// 2) row dots: xa[n] = x[n,:].att[:F]; ha[m] = he[m,:].att[F:]. Wave per row.
// ---------------------------------------------------------------------------
__global__ __launch_bounds__(256)
void rowdot_kernel(const float* __restrict__ x, const float* __restrict__ he,
                   const float* __restrict__ att, float* __restrict__ xa,
                   float* __restrict__ ha, int n_nodes, int n_edges) {
    const int row  = blockIdx.x * 8 + (threadIdx.x >> 5);
    const int lane = threadIdx.x & 31;
    const float* src;
    const float* a;
    float* dst;
    if (row < n_nodes) {
        src = x + (size_t)row * FDIM; a = att;        dst = xa + row;
    } else if (row < n_nodes + n_edges) {
        const int r = row - n_nodes;
        src = he + (size_t)r * FDIM;  a = att + FDIM; dst = ha + r;
    } else {
        return;
    }
    float s = 0.0f;
    #pragma unroll
    for (int f = lane; f < FDIM; f += 32) s += src[f] * a[f];
    #pragma unroll
    for (int off = 16; off > 0; off >>= 1) s += __shfl_xor(s, off, 32);
    if (lane == 0) *dst = s;
}

// ---------------------------------------------------------------------------
// 3) alpha logits + leaky(0.2) + segment max + degree counts
// ---------------------------------------------------------------------------
__global__ __launch_bounds__(256)
void alpha_kernel(const int* __restrict__ node_idx, const int* __restrict__ edge_idx,
                  const float* __restrict__ xa, const float* __restrict__ ha,
                  float* __restrict__ alpha, float* __restrict__ amax,
                  float* __restrict__ Dcnt, float* __restrict__ Bcnt, int ne) {
    const int e = blockIdx.x * 256 + threadIdx.x;
    if (e >= ne) return;
    const int n = node_idx[e];
    const int m = edge_idx[e];
    float v = xa[n] + ha[m];
    v = (v > 0.0f) ? v : 0.2f * v;
    alpha[e] = v;
    atomicMaxF32(&amax[n], v);
    atomicAdd(&Dcnt[n], 1.0f);
    atomicAdd(&Bcnt[m], 1.0f);
}

// ---------------------------------------------------------------------------
// 4) ex = exp(alpha - amax[node]); denom[node] += ex  (alpha overwritten)
// ---------------------------------------------------------------------------
__global__ __launch_bounds__(256)
void expdenom_kernel(const int* __restrict__ node_idx, const float* __restrict__ amax,
                     float* __restrict__ alpha, float* __restrict__ denom, int ne) {
    const int e = blockIdx.x * 256 + threadIdx.x;
    if (e >= ne) return;
    const int n = node_idx[e];
    const float ex = expf(alpha[e] - amax[n]);
    alpha[e] = ex;
    atomicAdd(&denom[n], ex);
}

// ---------------------------------------------------------------------------
// 5) edge_out[m,:] += (ex/denom[n]) * Binv[m] * x[n,:]   (wave per incidence)
// ---------------------------------------------------------------------------
__global__ __launch_bounds__(256)
void scatter_edge_kernel(const int* __restrict__ node_idx, const int* __restrict__ edge_idx,
                         const float* __restrict__ alpha, const float* __restrict__ denom,
                         const float* __restrict__ Bcnt, const float* __restrict__ x,
                         float* __restrict__ edge_out, int ne) {
    const int e = blockIdx.x * 8 + (threadIdx.x >> 5);
    if (e >= ne) return;
    const int lane = threadIdx.x & 31;
    const int n = node_idx[e];
    const int m = edge_idx[e];
    const float coef = alpha[e] / denom[n] / Bcnt[m];
    const float4* xs = (const float4*)(x + (size_t)n * FDIM);
    float* eo = edge_out + (size_t)m * FDIM;
    #pragma unroll
    for (int i = 0; i < 2; ++i) {
        const int vi = lane + i * 32;      // 64 float4 per row
        float4 v = xs[vi];
        const int base = vi * 4;
        atomicAdd(&eo[base + 0], coef * v.x);
        atomicAdd(&eo[base + 1], coef * v.y);
        atomicAdd(&eo[base + 2], coef * v.z);
        atomicAdd(&eo[base + 3], coef * v.w);
    }
}

// ---------------------------------------------------------------------------
// 6) out_acc[n,:] += (ex/denom[n]) * Dinv[n] * edge_out[m,:]
// ---------------------------------------------------------------------------
__global__ __launch_bounds__(256)
void scatter_node_kernel(const int* __restrict__ node_idx, const int* __restrict__ edge_idx,
                         const float* __restrict__ alpha, const float* __restrict__ denom,
                         const float* __restrict__ Dcnt, const float* __restrict__ edge_out,
                         float* __restrict__ out_acc, int ne) {
    const int e = blockIdx.x * 8 + (threadIdx.x >> 5);
    if (e >= ne) return;
    const int lane = threadIdx.x & 31;
    const int n = node_idx[e];
    const int m = edge_idx[e];
    const float coef = alpha[e] / denom[n] / Dcnt[n];
    const float4* es = (const float4*)(edge_out + (size_t)m * FDIM);
    float* no = out_acc + (size_t)n * FDIM;
    #pragma unroll
    for (int i = 0; i < 2; ++i) {
        const int vi = lane + i * 32;
        float4 v = es[vi];
        const int base = vi * 4;
        atomicAdd(&no[base + 0], coef * v.x);
        atomicAdd(&no[base + 1], coef * v.y);
        atomicAdd(&no[base + 2], coef * v.z);
        atomicAdd(&no[base + 3], coef * v.w);
    }
}

// ---------------------------------------------------------------------------
// 7) finalize in place: out = leaky(acc + bias, 0.01)
// ---------------------------------------------------------------------------
__global__ __launch_bounds__(256)
void finalize_kernel(float* __restrict__ out, const float* __restrict__ bias,
                     long long total) {
    long long stride = (long long)gridDim.x * blockDim.x;
    for (long long i = (long long)blockIdx.x * blockDim.x + threadIdx.x;
         i < total; i += stride) {
        float v = out[i] + bias[i & (FDIM - 1)];
        out[i] = (v > 0.0f) ? v : 0.01f * v;
    }
}

// ---------------------------------------------------------------------------
extern "C" void kernel_launch(void* const* d_in, const int* in_sizes, int n_in,
                              void* d_out, int out_size, void* d_ws, size_t ws_size,
                              hipStream_t stream) {
    const float* x    = (const float*)d_in[0];
    const float* hea  = (const float*)d_in[1];
    const float* W    = (const float*)d_in[2];
    const float* att  = (const float*)d_in[3];
    const float* bias = (const float*)d_in[4];
    const int* node_idx = (const int*)d_in[5];
    const int* edge_idx = (const int*)d_in[6];

    const int F = in_sizes[4];            // 256
    const int N = in_sizes[0] / F;        // 50000
    const int M = in_sizes[1] / F;        // 5000
    const int E = in_sizes[5];            // 200000

    float* out_acc = (float*)d_out;       // N*F accumulator, finalized in place

    // workspace carve-out (~12 MB)
    float* ws       = (float*)d_ws;
    float* he       = ws;                           // M*F
    float* edge_out = he + (size_t)M * F;           // M*F
    float* xa       = edge_out + (size_t)M * F;     // N
    float* ha       = xa + N;                       // M
    float* alpha    = ha + M;                       // E
    float* amax     = alpha + E;                    // N
    float* denom    = amax + N;                     // N
    float* Dcnt     = denom + N;                    // N
    float* Bcnt     = Dcnt + N;                     // M

    const long long NF = (long long)N * F;
    const long long MF = (long long)M * F;

    // 0) init accumulators
    {
        int blocks = (int)((NF + 255) / 256);
        if (blocks > 65535) blocks = 65535;
        init_kernel<<<blocks, 256, 0, stream>>>(out_acc, edge_out, amax, denom,
                                                Dcnt, Bcnt, NF, MF, N, M);
    }
    // 1) he = hea @ W  (WMMA fp32)
    {
        const int tilesM = (M + 15) / 16;           // 313
        const int tiles  = tilesM * (F / 16);       // 5008
        const int blocks = (tiles + 7) / 8;         // 8 waves/block
        he_gemm_kernel<<<blocks, 256, 0, stream>>>(hea, W, he, M);
    }
    // 2) row dots for attention logits
    {
        const int rows = N + M;
        rowdot_kernel<<<(rows + 7) / 8, 256, 0, stream>>>(x, he, att, xa, ha, N, M);
    }
    // 3) logits + leaky + segmax + counts
    alpha_kernel<<<(E + 255) / 256, 256, 0, stream>>>(node_idx, edge_idx, xa, ha,
                                                      alpha, amax, Dcnt, Bcnt, E);
    // 4) exp + denom
    expdenom_kernel<<<(E + 255) / 256, 256, 0, stream>>>(node_idx, amax, alpha, denom, E);
    // 5) node -> hyperedge scatter
    scatter_edge_kernel<<<(E + 7) / 8, 256, 0, stream>>>(node_idx, edge_idx, alpha,
                                                         denom, Bcnt, x, edge_out, E);
    // 6) hyperedge -> node scatter
    scatter_node_kernel<<<(E + 7) / 8, 256, 0, stream>>>(node_idx, edge_idx, alpha,
                                                         denom, Dcnt, edge_out, out_acc, E);
    // 7) bias + leaky(0.01) in place
    {
        int blocks = (int)((NF + 255) / 256);
        if (blocks > 65535) blocks = 65535;
        finalize_kernel<<<blocks, 256, 0, stream>>>(out_acc, bias, NF);
    }
}